// CUDAComputeGradientModule_59605556133985
// MI455X (gfx1250) — compile-verified
//
#include <hip/hip_runtime.h>
#include <math.h>

#define TILE 256
#define CLAMP_THR 0.9999999f

// Low 32 bits of a generic pointer to __shared__ data == LDS byte offset
// (shared aperture base has zero low 32 bits on gfx1250; verified in round-1 asm).
__device__ __forceinline__ unsigned lds_off(const float* p) {
    return (unsigned)(unsigned long long)p;
}

// CDNA5 async global->LDS dword copy (GVS mode). ASYNCcnt-tracked, bypasses VGPRs.
__device__ __forceinline__ void async_g2l_b32(unsigned lds_byte, const float* base,
                                              unsigned gbyte) {
    asm volatile("global_load_async_to_lds_b32 %0, %1, %2"
                 :: "v"(lds_byte), "v"(gbyte), "s"(base)
                 : "memory");
}

// CDNA5 async LDS->global dword copy (GVS mode). ASYNCcnt-tracked, bypasses VGPRs.
// Operand order mirrors global_store_b32: vaddr, vsrc(lds), saddr.
__device__ __forceinline__ void async_l2g_b32(float* base, unsigned gbyte,
                                              unsigned lds_byte) {
    asm volatile("global_store_async_from_lds_b32 %0, %1, %2"
                 :: "v"(gbyte), "v"(lds_byte), "s"(base)
                 : "memory");
}

__device__ __forceinline__ void wait_asynccnt0() {
    asm volatile("s_wait_asynccnt 0" ::: "memory");
}

__global__ __launch_bounds__(256) void geodesic_loss_grad_kernel(
    const float* __restrict__ R_rel, const float* __restrict__ R_w2c1,
    const float* __restrict__ R_w2c2, float* __restrict__ out,
    float* __restrict__ partial, int n)
{
    __shared__ float smem[3 * TILE * 9];   // 27,648 B: rel | w1 | w2 (reused for d1 | d2)
    __shared__ float wsum[8];

    const int tid   = threadIdx.x;
    const int blk   = blockIdx.x;
    const int base  = blk * TILE;                      // first element of tile
    const int count = (n - base) < TILE ? (n - base) : TILE;
    const int nf    = count * 9;                       // floats per array in tile
    const unsigned gbyte0 = (unsigned)base * 36u;      // byte offset of tile start

    float* s_rel = smem;
    float* s_w1  = smem + TILE * 9;
    float* s_w2  = smem + 2 * TILE * 9;

    // ---- coalesced async staging: 3 arrays x nf dwords, memory -> LDS ----
    for (int i = tid; i < nf; i += 256) {
        const unsigned gb = gbyte0 + (unsigned)i * 4u;
        async_g2l_b32(lds_off(s_rel + i), R_rel,  gb);
        async_g2l_b32(lds_off(s_w1  + i), R_w2c1, gb);
        async_g2l_b32(lds_off(s_w2  + i), R_w2c2, gb);
    }
    wait_asynccnt0();
    __syncthreads();

    // ---- per-element math (one element per thread; LDS reads stride-9 -> no bank conflict)
    float M[9], P[9];
    float lossv = 0.0f, g = 0.0f;
    if (tid < count) {
        float a[9], b1[9], b2[9];
        #pragma unroll
        for (int m = 0; m < 9; ++m) {
            a[m]  = s_rel[tid * 9 + m];   // R_rel[r][c] at r*3+c
            b1[m] = s_w1 [tid * 9 + m];
            b2[m] = s_w2 [tid * 9 + m];
        }
        float tr = 0.0f;
        #pragma unroll
        for (int i = 0; i < 3; ++i) {
            #pragma unroll
            for (int k = 0; k < 3; ++k) {
                // M = rel^T * w2
                const float m_ik = a[0 + i] * b2[0 + k]
                                 + a[3 + i] * b2[3 + k]
                                 + a[6 + i] * b2[6 + k];
                M[i * 3 + k] = m_ik;
                tr += m_ik * b1[i * 3 + k];
                // P = rel * w1
                P[i * 3 + k] = a[i * 3 + 0] * b1[0 + k]
                             + a[i * 3 + 1] * b1[3 + k]
                             + a[i * 3 + 2] * b1[6 + k];
            }
        }
        const float c  = 0.5f * (tr - 1.0f);
        const float cc = fminf(fmaxf(c, -CLAMP_THR), CLAMP_THR);
        lossv = acosf(cc);
        g = (fabsf(c) < CLAMP_THR) ? (-0.5f / sqrtf(1.0f - cc * cc)) : 0.0f;
    }
    __syncthreads();   // all LDS input reads done before buffer reuse

    // ---- stage gradients into LDS (reuse buffers) ----
    if (tid < count) {
        #pragma unroll
        for (int m = 0; m < 9; ++m) {
            s_rel[tid * 9 + m] = g * M[m];   // d_R_w2c1
            s_w1 [tid * 9 + m] = g * P[m];   // d_R_w2c2
        }
    }
    __syncthreads();

    // ---- coalesced async stores: LDS -> memory, no VGPR round-trip ----
    // d_out layout: [loss(1) | d_R_w2c1(9n) | d_R_w2c2(9n)]; byte offsets < 2^31.
    {
        const unsigned o1b0 = 4u + (unsigned)base * 36u;                       // +1 float header
        const unsigned o2b0 = 4u + (unsigned)n * 36u + (unsigned)base * 36u;
        for (int i = tid; i < nf; i += 256) {
            const unsigned ib = (unsigned)i * 4u;
            async_l2g_b32(out, o1b0 + ib, lds_off(s_rel + i));
            async_l2g_b32(out, o2b0 + ib, lds_off(s_w1  + i));
        }
        // No explicit wait needed: S_ENDPGM performs an implicit wait-idle, and the
        // remaining code only touches wsum[] (disjoint LDS) and out[0]/partial[].
    }

    // ---- loss reduction: wave32 shuffle tree -> 8 wave sums -> block sum ----
    float s = lossv;
    #pragma unroll
    for (int d = 16; d > 0; d >>= 1) s += __shfl_down(s, d, 32);
    if ((tid & 31) == 0) wsum[tid >> 5] = s;
    __syncthreads();
    if (tid == 0) {
        float t = 0.0f;
        #pragma unroll
        for (int w = 0; w < 8; ++w) t += wsum[w];
        if (partial) partial[blk] = t;          // deterministic two-pass path
        else         atomicAdd(out, t);         // fallback if workspace too small
    }
}

__global__ __launch_bounds__(256) void loss_reduce_kernel(
    const float* __restrict__ partial, int nb, float* __restrict__ out)
{
    __shared__ float sm[256];
    float s = 0.0f;
    for (int i = threadIdx.x; i < nb; i += 256) s += partial[i];  // fixed order
    sm[threadIdx.x] = s;
    __syncthreads();
    #pragma unroll
    for (int d = 128; d > 0; d >>= 1) {
        if (threadIdx.x < d) sm[threadIdx.x] += sm[threadIdx.x + d];
        __syncthreads();
    }
    if (threadIdx.x == 0) out[0] = sm[0];
}

__global__ void zero_loss_kernel(float* out) {
    if (threadIdx.x == 0) out[0] = 0.0f;
}

extern "C" void kernel_launch(void* const* d_in, const int* in_sizes, int n_in,
                              void* d_out, int out_size, void* d_ws, size_t ws_size,
                              hipStream_t stream) {
    (void)n_in; (void)out_size;
    const float* R_rel  = (const float*)d_in[0];
    const float* R_w2c1 = (const float*)d_in[1];
    const float* R_w2c2 = (const float*)d_in[2];
    float* out = (float*)d_out;

    const int n  = in_sizes[0] / 9;
    const int nb = (n + TILE - 1) / TILE;

    const bool use_ws = ws_size >= (size_t)nb * sizeof(float);
    float* partial = use_ws ? (float*)d_ws : nullptr;

    if (!use_ws) zero_loss_kernel<<<1, 64, 0, stream>>>(out);
    geodesic_loss_grad_kernel<<<nb, 256, 0, stream>>>(R_rel, R_w2c1, R_w2c2,
                                                      out, partial, n);
    if (use_ws) loss_reduce_kernel<<<1, 256, 0, stream>>>(partial, nb, out);
}